// FusionconBlock_38869454029173
// MI455X (gfx1250) — compile-verified
//
#include <hip/hip_runtime.h>

typedef __attribute__((ext_vector_type(2))) float v2f;
typedef __attribute__((ext_vector_type(8))) float v8f;

#define HW   16384   // H*W = 128*128
#define CPH  16      // channels per head
#define CTOT 64      // total channels
#define NBH  64      // B * heads = 16*4

// ---------------------------------------------------------------------------
// Kernel 1: per (b,h) block computes
//   E = norm_rows(X) @ norm_rows(Y)^T   (16x16, K=16384) via V_WMMA_F32_16X16X4_F32
// with row L2-norms fused into the same streaming loop, then both softmaxes
// (over d, and over the transposed axis), temperatures folded in, and the two
// 16x16 weight matrices written to workspace.
// ---------------------------------------------------------------------------
__global__ __launch_bounds__(512)
void energy_softmax_kernel(const float* __restrict__ x,
                           const float* __restrict__ y,
                           const float* __restrict__ t1p,
                           const float* __restrict__ t2p,
                           float* __restrict__ wsw) {
  __shared__ float e[256];
  __shared__ float a1[256];
  __shared__ float a2[256];
  __shared__ float nxs[16];
  __shared__ float nys[16];

  const int tid = threadIdx.x;
  const int bh  = blockIdx.x;                 // b*4 + h
  const float* __restrict__ X = x + (size_t)bh * CPH * HW;
  const float* __restrict__ Y = y + (size_t)bh * CPH * HW;

  if (tid < 256)                 e[tid]        = 0.0f;
  else if (tid < 272)            nxs[tid-256]  = 0.0f;
  else if (tid < 288)            nys[tid-272]  = 0.0f;
  __syncthreads();

  const int lane = tid & 31;
  const int wave = tid >> 5;                  // 16 waves / block
  const int m    = lane & 15;                 // matrix row (A) / col (B)
  const int half = lane >> 4;                 // K-half select per ISA layout

  v8f  acc = {};
  float nx = 0.0f, ny = 0.0f;

  const int kbeg = wave * (HW / 16);
  const int kend = kbeg + (HW / 16);
  const float* __restrict__ xrow = X + (size_t)m * HW;
  const float* __restrict__ yrow = Y + (size_t)m * HW;

#pragma unroll 4
  for (int kk = kbeg; kk < kend; kk += 4) {
    const int col = kk + half * 2;
    // A 16x4 f32 fragment: lanes 0-15 hold K=0,1; lanes 16-31 hold K=2,3
    v2f a = *(const v2f*)(xrow + col);
    // B 4x16 f32 fragment (Y^T): same per-lane pattern with n = lane&15
    v2f b = *(const v2f*)(yrow + col);
    nx += a.x * a.x + a.y * a.y;
    ny += b.x * b.x + b.y * b.y;
    // (neg_a, A, neg_b, B, c_mod, C, reuse_a, reuse_b)
    acc = __builtin_amdgcn_wmma_f32_16x16x4_f32(false, a, false, b,
                                                (short)0, acc, false, false);
  }

  // Combine the 16 per-wave partial tiles + norms in LDS (ds_add_f32).
  atomicAdd(&nxs[m], nx);
  atomicAdd(&nys[m], ny);
#pragma unroll
  for (int r = 0; r < 8; ++r) {
    // C/D layout: VGPR r, lanes 0-15 -> M=r, N=lane; lanes 16-31 -> M=r+8
    const int M = r + 8 * half;
    atomicAdd(&e[M * 16 + m], acc[r]);
  }
  __syncthreads();

  // Normalize: E[c,d] / (max(||Xc||,eps) * max(||Yd||,eps))
  if (tid < 256) {
    const int M = tid >> 4, N = tid & 15;
    const float dn = fmaxf(sqrtf(nxs[M]), 1e-12f) * fmaxf(sqrtf(nys[N]), 1e-12f);
    e[tid] = e[tid] / dn;
  }
  __syncthreads();

  const float t1 = t1p[0];
  const float t2 = t2p[0];

  if (tid < 16) {
    // attn1[c,d] = softmax over d of E[c,:]  ->  W1[c*16+d] = t1 * attn1[c,d]
    const int c = tid;
    float mx = -3.4e38f;
#pragma unroll
    for (int d = 0; d < 16; ++d) mx = fmaxf(mx, e[c * 16 + d]);
    float s = 0.0f, ex[16];
#pragma unroll
    for (int d = 0; d < 16; ++d) { ex[d] = __expf(e[c * 16 + d] - mx); s += ex[d]; }
    const float inv = t1 / s;
#pragma unroll
    for (int d = 0; d < 16; ++d) a1[c * 16 + d] = ex[d] * inv;
  } else if (tid < 32) {
    // attn2[c,d] = exp(E[d,c]) / sum_{d'} exp(E[d',c])  (column softmax of E)
    // store W2[d*16+c] = t2 * attn2[c,d] so kernel 2 reads contiguously per d.
    const int c = tid - 16;
    float mx = -3.4e38f;
#pragma unroll
    for (int d = 0; d < 16; ++d) mx = fmaxf(mx, e[d * 16 + c]);
    float s = 0.0f, ex[16];
#pragma unroll
    for (int d = 0; d < 16; ++d) { ex[d] = __expf(e[d * 16 + c] - mx); s += ex[d]; }
    const float inv = t2 / s;
#pragma unroll
    for (int d = 0; d < 16; ++d) a2[d * 16 + c] = ex[d] * inv;
  }
  __syncthreads();

  if (tid < 256)      wsw[(size_t)bh * 512 + tid]       = a1[tid];
  else if (tid < 512) wsw[(size_t)bh * 512 + tid]       = a2[tid - 256];
}

// ---------------------------------------------------------------------------
// Kernel 2: streaming apply + residual + concat-store.
//   y_out[d,k] = X[d,k] + sum_c W1[c,d] * Y[c,k]      (W1 pre-scaled by t1)
//   x_out[d,k] = Y[d,k] + sum_c W2[d,c] * X[c,k]      (W2 pre-scaled by t2)
// One thread owns 4 consecutive pixels across all 16 channels of one head.
// ---------------------------------------------------------------------------
__global__ __launch_bounds__(256)
void apply_kernel(const float* __restrict__ x,
                  const float* __restrict__ y,
                  const float* __restrict__ wsw,
                  float* __restrict__ out) {
  __shared__ float w1[256];
  __shared__ float w2[256];

  const int tid = threadIdx.x;
  const int bh  = blockIdx.x >> 4;    // 0..63
  const int kg  = blockIdx.x & 15;    // pixel-group within head
  const int b   = bh >> 2;
  const int h   = bh & 3;

  w1[tid] = wsw[(size_t)bh * 512 + tid];
  w2[tid] = wsw[(size_t)bh * 512 + 256 + tid];
  __syncthreads();

  const int k = (kg * 256 + tid) * 4;
  const float* __restrict__ Xb = x + (size_t)bh * CPH * HW + k;
  const float* __restrict__ Yb = y + (size_t)bh * CPH * HW + k;

  float4 xv[16], yv[16];
#pragma unroll
  for (int c = 0; c < 16; ++c) {
    xv[c] = *(const float4*)(Xb + (size_t)c * HW);
    yv[c] = *(const float4*)(Yb + (size_t)c * HW);
  }

  float* __restrict__ outX = out + ((size_t)b * 2 * CTOT + h * CPH) * HW + k;
  float* __restrict__ outY = out + ((size_t)b * 2 * CTOT + CTOT + h * CPH) * HW + k;

#pragma unroll
  for (int d = 0; d < 16; ++d) {
    float4 sy = make_float4(0.f, 0.f, 0.f, 0.f);
    float4 sx = make_float4(0.f, 0.f, 0.f, 0.f);
#pragma unroll
    for (int c = 0; c < 16; ++c) {
      const float wy = w1[c * 16 + d];   // LDS broadcast (uniform addr)
      const float wx = w2[d * 16 + c];
      sy.x += wy * yv[c].x; sy.y += wy * yv[c].y;
      sy.z += wy * yv[c].z; sy.w += wy * yv[c].w;
      sx.x += wx * xv[c].x; sx.y += wx * xv[c].y;
      sx.z += wx * xv[c].z; sx.w += wx * xv[c].w;
    }
    float4 oy, ox;
    oy.x = sy.x + xv[d].x; oy.y = sy.y + xv[d].y;
    oy.z = sy.z + xv[d].z; oy.w = sy.w + xv[d].w;
    ox.x = sx.x + yv[d].x; ox.y = sx.y + yv[d].y;
    ox.z = sx.z + yv[d].z; ox.w = sx.w + yv[d].w;
    *(float4*)(outY + (size_t)d * HW) = oy;
    *(float4*)(outX + (size_t)d * HW) = ox;
  }
}

extern "C" void kernel_launch(void* const* d_in, const int* in_sizes, int n_in,
                              void* d_out, int out_size, void* d_ws, size_t ws_size,
                              hipStream_t stream) {
  const float* x  = (const float*)d_in[0];
  const float* y  = (const float*)d_in[1];
  const float* t1 = (const float*)d_in[2];
  const float* t2 = (const float*)d_in[3];
  // d_in[4] = num_heads (compile-time constant 4 here)
  float* out = (float*)d_out;
  float* ws  = (float*)d_ws;   // needs 64*512*4 = 128 KB

  energy_softmax_kernel<<<NBH, 512, 0, stream>>>(x, y, t1, t2, ws);
  apply_kernel<<<NBH * 16, 256, 0, stream>>>(x, y, ws, out);
}